// Mamba2TextEncoder_72060961292325
// MI455X (gfx1250) — compile-verified
//
#include <hip/hip_runtime.h>
#include <hip/hip_bf16.h>
#include <math.h>

// ---------------- problem constants (match reference) ----------------
enum : int {
    KV = 49408, KLMAX = 512, KDMODEL = 768, KNLAYER = 4, KNSTATE = 16,
    KDINNER = 1536, KDTRANK = 48, KDCONV = 4, KB = 4, KT = 256,
    KROWS = KB * KT,                     // 1024 (M dimension of all GEMMs)
    KXDBL = KDTRANK + 2 * KNSTATE        // 80
};
#define KEPS 1e-5f

typedef __attribute__((ext_vector_type(2))) float v2f;
typedef __attribute__((ext_vector_type(8))) float v8f;
typedef int v4i __attribute__((vector_size(4 * sizeof(int))));
typedef __attribute__((address_space(1))) v4i* as1_v4i_p;   // global
typedef __attribute__((address_space(3))) v4i* as3_v4i_p;   // LDS

#if __has_builtin(__builtin_amdgcn_global_load_async_to_lds_b128) && \
    __has_builtin(__builtin_amdgcn_s_wait_asynccnt)
#define HAVE_ASYNC_LDS 1
#else
#define HAVE_ASYNC_LDS 0
#endif

// ---------------------------------------------------------------------
// Generic fp32 WMMA GEMM:  C(M,N) = A(M,K; lda) * W(N,K; ldw)^T
// Register-blocked: each wave32 computes 1 M-tile x NB N-tiles, reusing
// the A fragment NB times per K-step (V_WMMA_F32_16X16X4_F32, full fp32).
// All NB B fragments are loaded into distinct locals before the WMMA
// burst so the backend can clause the loads and stagger s_wait_loadcnt.
// epilogue: 0 = store, 1 = store softplus(acc + bias[n]), 2 = C += acc
// A-frag layout (ISA 7.12.2, 32-bit A 16x4): lanes 0-15 hold K={0,1},
// lanes 16-31 hold K={2,3}; lane%16 = row. B mirrors with lane%16 = col.
// C/D layout: VGPR i -> row = i + 8*(lane/16), col = lane%16.
// ---------------------------------------------------------------------
template <int NB>
__global__ void wmma_gemm_nt(const float* __restrict__ A,
                             const float* __restrict__ W,
                             float* __restrict__ C,
                             const float* __restrict__ bias,
                             int M, int N, int K,
                             int lda, int ldw, int ldc, int epilogue)
{
    const int wave    = (blockIdx.x * blockDim.x + threadIdx.x) >> 5;
    const int lane    = threadIdx.x & 31;
    const int ngroups = (N >> 4) / NB;           // N-tile groups per row panel
    const int tm  = (wave / ngroups) << 4;
    const int tn0 = (wave % ngroups) * (NB << 4);
    if (tm >= M) return;

    const int half = lane >> 4;                  // 0 -> K pair {0,1}, 1 -> {2,3}
    const int l16  = lane & 15;

    const float* __restrict__ arow = A + (size_t)(tm + l16) * lda + half * 2;
    const float* wrow[NB];
#pragma unroll
    for (int nb = 0; nb < NB; ++nb)
        wrow[nb] = W + (size_t)(tn0 + (nb << 4) + l16) * ldw + half * 2;

    v8f zero = {};
    v8f acc[NB];
#pragma unroll
    for (int nb = 0; nb < NB; ++nb) acc[nb] = zero;

    for (int k = 0; k < K; k += 4) {
        // issue all fragment loads first (clause + staggered waits) ...
        v2f a = *(const v2f*)(arow + k);
        v2f bfr[NB];
#pragma unroll
        for (int nb = 0; nb < NB; ++nb)
            bfr[nb] = *(const v2f*)(wrow[nb] + k);
        // stream-ahead hint on the weight panel (-> global_prefetch_b8)
        __builtin_prefetch(wrow[0] + k + 512, 0, 1);
        // ... then the WMMA burst
#pragma unroll
        for (int nb = 0; nb < NB; ++nb) {
            acc[nb] = __builtin_amdgcn_wmma_f32_16x16x4_f32(
                /*neg_a=*/false, a, /*neg_b=*/false, bfr[nb],
                /*c_mod=*/(short)0, acc[nb],
                /*reuse_a=*/false, /*reuse_b=*/false);
        }
    }

    const int crow0 = tm + half * 8;
#pragma unroll
    for (int nb = 0; nb < NB; ++nb) {
        const int ccol = tn0 + (nb << 4) + l16;
#pragma unroll
        for (int i = 0; i < 8; ++i) {
            float v = acc[nb][i];
            float* cp = C + (size_t)(crow0 + i) * ldc + ccol;
            if (epilogue == 1) {
                float x = v + bias[ccol];
                *cp = (x > 20.f) ? x : log1pf(__expf(x));    // softplus
            } else if (epilogue == 2) {
                *cp += v;                                    // residual accumulate
            } else {
                *cp = v;
            }
        }
    }
}

// ---------------- embedding gather + positional add ------------------
__global__ void embed_kernel(const int* __restrict__ ids,
                             const float* __restrict__ emb,
                             const float* __restrict__ pos,
                             float* __restrict__ x)
{
    int idx = blockIdx.x * blockDim.x + threadIdx.x;
    if (idx >= KROWS * KDMODEL) return;
    int c   = idx % KDMODEL;
    int row = idx / KDMODEL;           // b*T + t
    int t   = row % KT;
    int id  = ids[row];
    x[idx] = emb[(size_t)id * KDMODEL + c] + pos[(size_t)t * KDMODEL + c];
}

// -------- causal depthwise conv (K=4) + bias + SiLU over xz[:, :D] ---
__global__ void conv_silu_kernel(const float* __restrict__ xz,   // (ROWS, 2*D)
                                 const float* __restrict__ cw,   // (D, 4)
                                 const float* __restrict__ cb,   // (D,)
                                 float* __restrict__ xi)         // (ROWS, D)
{
    int idx = blockIdx.x * blockDim.x + threadIdx.x;
    if (idx >= KROWS * KDINNER) return;
    int d   = idx % KDINNER;
    int row = idx / KDINNER;
    int t   = row % KT;
    int b   = row / KT;

    float acc = cb[d];
#pragma unroll
    for (int k = 0; k < KDCONV; ++k) {
        int tt = t - (KDCONV - 1) + k;
        if (tt >= 0) {
            acc += cw[d * KDCONV + k] *
                   xz[(size_t)(b * KT + tt) * (2 * KDINNER) + d];
        }
    }
    xi[idx] = acc / (1.f + __expf(-acc));                // SiLU
}

// ---------------- selective scan + D-skip + SiLU(z) gate -------------
// One lane per (b, d) channel; 16-state h kept in registers. The per-step
// B/C vectors (32 contiguous, 16B-aligned floats shared by the whole
// block) are moved global->LDS with GLOBAL_LOAD_ASYNC_TO_LDS_B128
// (ASYNCcnt-tracked), 8 lanes x b128 = 128 bytes, then s_wait_asynccnt 0.
__global__ void scan_kernel(const float* __restrict__ xdbl,  // (ROWS, 80)
                            const float* __restrict__ dtb,   // (ROWS, D) softplus'd
                            const float* __restrict__ xi,    // (ROWS, D)
                            const float* __restrict__ xz,    // (ROWS, 2D) (z = [:,D:])
                            const float* __restrict__ A_log, // (D, 16)
                            const float* __restrict__ Dskip, // (D,)
                            float* __restrict__ y)           // (ROWS, D)
{
    const int chunks = KDINNER / 256;              // 6
    const int b = blockIdx.x / chunks;
    const int d = (blockIdx.x % chunks) * 256 + threadIdx.x;

    float Av[KNSTATE];
#pragma unroll
    for (int n = 0; n < KNSTATE; ++n)
        Av[n] = -__expf(A_log[(size_t)d * KNSTATE + n]);
    const float Dv = Dskip[d];

    float h[KNSTATE];
#pragma unroll
    for (int n = 0; n < KNSTATE; ++n) h[n] = 0.f;

    __shared__ __align__(16) float sBC[2 * KNSTATE];   // [0:16)=B, [16:32)=C

    for (int t = 0; t < KT; ++t) {
        const size_t row = (size_t)b * KT + t;

#if HAVE_ASYNC_LDS
        if (threadIdx.x < 8) {
            const float* g = xdbl + row * KXDBL + KDTRANK + threadIdx.x * 4;
            __builtin_amdgcn_global_load_async_to_lds_b128(
                (as1_v4i_p)g,
                (as3_v4i_p)(sBC + threadIdx.x * 4),
                0, 0);
        }
        __builtin_amdgcn_s_wait_asynccnt(0);
        __syncthreads();
#else
        if (threadIdx.x < 2 * KNSTATE)
            sBC[threadIdx.x] = xdbl[row * KXDBL + KDTRANK + threadIdx.x];
        __syncthreads();
#endif

        const float dtv = dtb[row * KDINNER + d];
        const float xiv = xi [row * KDINNER + d];
        const float zv  = xz [row * (2 * KDINNER) + KDINNER + d];
        const float dx  = dtv * xiv;

        float yv = Dv * xiv;
#pragma unroll
        for (int n = 0; n < KNSTATE; ++n) {
            h[n] = __expf(dtv * Av[n]) * h[n] + dx * sBC[n];
            yv  += h[n] * sBC[KNSTATE + n];
        }
        y[row * KDINNER + d] = yv * (zv / (1.f + __expf(-zv)));  // * SiLU(z)
        __syncthreads();
    }
}

// ---------------------------- RMSNorm --------------------------------
__global__ void rmsnorm_kernel(const float* __restrict__ x,
                               const float* __restrict__ w,
                               float* __restrict__ out)
{
    __shared__ float red[256];
    const size_t row = blockIdx.x;               // 0 .. ROWS-1
    const int tid = threadIdx.x;

    float s = 0.f;
    for (int c = tid; c < KDMODEL; c += 256) {
        float v = x[row * KDMODEL + c];
        s += v * v;
    }
    red[tid] = s;
    __syncthreads();
    for (int off = 128; off > 0; off >>= 1) {
        if (tid < off) red[tid] += red[tid + off];
        __syncthreads();
    }
    const float scale = rsqrtf(red[0] / (float)KDMODEL + KEPS);
    for (int c = tid; c < KDMODEL; c += 256) {
        out[row * KDMODEL + c] = x[row * KDMODEL + c] * scale * w[c];
    }
}

// ---------------------------------------------------------------------
static inline int gemm_blocks(int M, int N, int NB)
{
    int waves = (M >> 4) * ((N >> 4) / NB);
    return (waves + 7) / 8;                      // 8 waves (256 thr) per block
}

extern "C" void kernel_launch(void* const* d_in, const int* in_sizes, int n_in,
                              void* d_out, int out_size, void* d_ws, size_t ws_size,
                              hipStream_t stream)
{
    (void)in_sizes; (void)n_in; (void)out_size; (void)ws_size;

    const int*   ids   = (const int*)  d_in[0];
    const float* emb   = (const float*)d_in[1];
    const float* pos   = (const float*)d_in[2];
    const float* ipw   = (const float*)d_in[3];   // (L, 3072, 768)
    const float* cw    = (const float*)d_in[4];   // (L, 1536, 4)
    const float* cb    = (const float*)d_in[5];   // (L, 1536)
    const float* xpw   = (const float*)d_in[6];   // (L, 80, 1536)
    const float* dpw   = (const float*)d_in[7];   // (L, 1536, 48)
    const float* dpb   = (const float*)d_in[8];   // (L, 1536)
    const float* A_log = (const float*)d_in[9];   // (L, 1536, 16)
    const float* Dsk   = (const float*)d_in[10];  // (L, 1536)
    const float* opw   = (const float*)d_in[11];  // (L, 768, 1536)
    const float* nw    = (const float*)d_in[12];  // (768,)
    float* out = (float*)d_out;

    // ---- workspace carve-up (floats) ----
    float* ws   = (float*)d_ws;
    float* xbuf = ws;                                     // ROWS*768
    float* xz   = xbuf + (size_t)KROWS * KDMODEL;         // ROWS*3072
    float* xi   = xz   + (size_t)KROWS * 2 * KDINNER;     // ROWS*1536
    float* xdbl = xi   + (size_t)KROWS * KDINNER;         // ROWS*80
    float* dtb  = xdbl + (size_t)KROWS * KXDBL;           // ROWS*1536
    float* yb   = dtb  + (size_t)KROWS * KDINNER;         // ROWS*1536

    // ---- embedding + positional ----
    {
        int n = KROWS * KDMODEL;
        embed_kernel<<<(n + 255) / 256, 256, 0, stream>>>(ids, emb, pos, xbuf);
    }

    // ---- layers ----
    for (int l = 0; l < KNLAYER; ++l) {
        const float* ipw_l = ipw   + (size_t)l * (2 * KDINNER) * KDMODEL;
        const float* cw_l  = cw    + (size_t)l * KDINNER * KDCONV;
        const float* cb_l  = cb    + (size_t)l * KDINNER;
        const float* xpw_l = xpw   + (size_t)l * KXDBL * KDINNER;
        const float* dpw_l = dpw   + (size_t)l * KDINNER * KDTRANK;
        const float* dpb_l = dpb   + (size_t)l * KDINNER;
        const float* Al_l  = A_log + (size_t)l * KDINNER * KNSTATE;
        const float* Ds_l  = Dsk   + (size_t)l * KDINNER;
        const float* opw_l = opw   + (size_t)l * KDMODEL * KDINNER;

        // in_proj: xz(1024,3072) = x(1024,768) * ipw^T     (192 N-tiles, NB=4)
        wmma_gemm_nt<4><<<gemm_blocks(KROWS, 2 * KDINNER, 4), 256, 0, stream>>>(
            xbuf, ipw_l, xz, nullptr,
            KROWS, 2 * KDINNER, KDMODEL, KDMODEL, KDMODEL, 2 * KDINNER, 0);

        // causal depthwise conv + SiLU on xz[:, :1536] -> xi
        {
            int n = KROWS * KDINNER;
            conv_silu_kernel<<<(n + 255) / 256, 256, 0, stream>>>(xz, cw_l, cb_l, xi);
        }

        // x_proj: xdbl(1024,80) = xi(1024,1536) * xpw^T    (5 N-tiles, NB=5)
        wmma_gemm_nt<5><<<gemm_blocks(KROWS, KXDBL, 5), 256, 0, stream>>>(
            xi, xpw_l, xdbl, nullptr,
            KROWS, KXDBL, KDINNER, KDINNER, KDINNER, KXDBL, 0);

        // dt_proj + softplus: dtb(1024,1536) = softplus(xdbl[:, :48]*dpw^T + dpb)
        wmma_gemm_nt<4><<<gemm_blocks(KROWS, KDINNER, 4), 256, 0, stream>>>(
            xdbl, dpw_l, dtb, dpb_l,
            KROWS, KDINNER, KDTRANK, KXDBL, KDTRANK, KDINNER, 1);

        // selective scan + D-skip + SiLU(z) gate
        scan_kernel<<<KB * (KDINNER / 256), 256, 0, stream>>>(
            xdbl, dtb, xi, xz, Al_l, Ds_l, yb);

        // out_proj with residual accumulate: x += y * opw^T (48 N-tiles, NB=4)
        wmma_gemm_nt<4><<<gemm_blocks(KROWS, KDMODEL, 4), 256, 0, stream>>>(
            yb, opw_l, xbuf, nullptr,
            KROWS, KDMODEL, KDINNER, KDINNER, KDINNER, KDMODEL, 2);
    }

    // ---- final RMSNorm -> d_out ----
    rmsnorm_kernel<<<KROWS, 256, 0, stream>>>(xbuf, nw, out);
}